// GraphQNN_65481071410812
// MI455X (gfx1250) — compile-verified
//
#include <hip/hip_runtime.h>
#include <hip/hip_bf16.h>
#include <math.h>

// ---------------------------------------------------------------------------
// GraphQNN on MI455X (gfx1250, wave32).
//   out[0..22] = h = tanh(W @ x)            -> V_WMMA_F32_16X16X4_F32 matvec
//   out[23]    = <Z_0> of the 23-qubit circuit, computed exactly via the
//                2-qubit backward light cone of Z_0 (see analysis).
// ---------------------------------------------------------------------------

typedef __attribute__((ext_vector_type(2))) float v2f;
typedef __attribute__((ext_vector_type(8))) float v8f;

#define QNN_NQ 23
#define QNN_IN_DIM 1024

// ---------------- Kernel 1: h = tanh(W @ x) via f32 WMMA -------------------
// One block of 64 threads = 2 waves. Wave w computes rows [16w, 16w+16).
// A(16x4) = W tile, B(4x16) = x chunk replicated across all N columns, so
// every column of D holds the row dot-products (layout-permutation robust).
__global__ __launch_bounds__(64) void qnn_matvec_wmma(const float* __restrict__ x,
                                                      const float* __restrict__ W,
                                                      float* __restrict__ out) {
  const int lane = threadIdx.x & 31;
  const int wave = threadIdx.x >> 5;
  const int m    = lane & 15;   // M row within tile (A) / N col (B)
  const int hi   = lane >> 4;   // lane half: selects K pair {2h, 2h+1}
  const int base = wave * 16;   // first output row for this wave

  int row  = base + m;
  int rowc = (row < QNN_NQ) ? row : (QNN_NQ - 1);   // clamp loads; writes guarded
  const float* __restrict__ wrow = W + (size_t)rowc * QNN_IN_DIM;

  v8f acc = {};
  #pragma unroll 4
  for (int k0 = 0; k0 < QNN_IN_DIM; k0 += 4) {
    const int k = k0 + 2 * hi;                 // K slot = vgpr + 2*half (A & B alike)
    v2f a; a.x = wrow[k]; a.y = wrow[k + 1];   // A[m][k], A[m][k+1]
    v2f b; b.x = x[k];    b.y = x[k + 1];      // B[k][n] = x[k] for all n
    acc = __builtin_amdgcn_wmma_f32_16x16x4_f32(
        /*neg_a=*/false, a, /*neg_b=*/false, b,
        /*c_mod=*/(short)0, acc, /*reuse_a=*/false, /*reuse_b=*/false);
  }

  // D layout: VGPR j -> row base+j (lanes 0-15) / base+8+j (lanes 16-31),
  // replicated across N. Lanes 0 and 16 each publish 8 rows.
  if (m == 0) {
    const int r0 = base + 8 * hi;
    #pragma unroll
    for (int j = 0; j < 8; ++j) {
      const int r = r0 + j;
      if (r < QNN_NQ) out[r] = tanhf(acc[j]);
    }
  }
}

// ---------------- Kernel 2: exact <Z_0> via 2-qubit light cone -------------
struct C2 { double re, im; };
__device__ __forceinline__ C2 cmul(C2 a, C2 b) {
  return { a.re * b.re - a.im * b.im, a.re * b.im + a.im * b.re };
}
__device__ __forceinline__ C2 csub(C2 a, C2 b) { return { a.re - b.re, a.im - b.im }; }
__device__ __forceinline__ C2 cconj(C2 a)      { return { a.re, -a.im }; }

// U = RZ(b) * RX(a)   (RX applied first, matching the reference)
__device__ __forceinline__ void make_u(double a, double b, C2 U[2][2]) {
  const double c = cos(a * 0.5), s = sin(a * 0.5);
  const C2 em = { cos(b * 0.5), -sin(b * 0.5) };  // e^{-ib/2}
  const C2 ep = { em.re, -em.im };                // e^{+ib/2}
  const C2 rx00 = { c, 0.0 }, rx01 = { 0.0, -s };
  U[0][0] = cmul(em, rx00);  U[0][1] = cmul(em, rx01);
  U[1][0] = cmul(ep, rx01);  U[1][1] = cmul(ep, rx00);
}

__global__ void qnn_expval(const float* __restrict__ q_params,
                           float* __restrict__ out) {
  if (threadIdx.x != 0 || blockIdx.x != 0) return;

  const double h0 = (double)out[0];
  const double h1 = (double)out[1];

  C2 U10[2][2], U11[2][2], U20[2][2];
  make_u((double)q_params[0], (double)q_params[1], U10);                 // layer0, q0
  make_u((double)q_params[2], (double)q_params[3], U11);                 // layer0, q1
  make_u((double)q_params[2 * QNN_NQ], (double)q_params[2 * QNN_NQ + 1], // layer1, q0
         U20);

  // O = U20^dagger Z U20   (Hermitian 2x2)
  C2 O[2][2];
  #pragma unroll
  for (int i = 0; i < 2; ++i)
    #pragma unroll
    for (int j = 0; j < 2; ++j)
      O[i][j] = csub(cmul(cconj(U20[0][i]), U20[0][j]),
                     cmul(cconj(U20[1][i]), U20[1][j]));

  // u = U10 * (cos h0, sin h0),  r = U11 * (cos h1, sin h1)
  const double c0 = cos(h0), s0 = sin(h0), c1 = cos(h1), s1 = sin(h1);
  C2 u[2], r[2];
  #pragma unroll
  for (int i = 0; i < 2; ++i) {
    u[i] = { U10[i][0].re * c0 + U10[i][1].re * s0,
             U10[i][0].im * c0 + U10[i][1].im * s0 };
    r[i] = { U11[i][0].re * c1 + U11[i][1].re * s1,
             U11[i][0].im * c1 + U11[i][1].im * s1 };
  }

  // w[b1][b0] = (r (x) u)[b1 ^ b0][b0]   (CNOT with control q0, target q1)
  C2 w[2][2];
  w[0][0] = cmul(r[0], u[0]);
  w[0][1] = cmul(r[1], u[1]);
  w[1][0] = cmul(r[1], u[0]);
  w[1][1] = cmul(r[0], u[1]);

  // exp = sum_{b1} w[b1]^dagger O w[b1]
  double ev = 0.0;
  #pragma unroll
  for (int b1 = 0; b1 < 2; ++b1) {
    const C2 wa = w[b1][0], wb = w[b1][1];
    const C2 t  = cmul(O[0][1], wb);
    ev += O[0][0].re * (wa.re * wa.re + wa.im * wa.im)
        + O[1][1].re * (wb.re * wb.re + wb.im * wb.im)
        + 2.0 * (wa.re * t.re + wa.im * t.im);
  }
  out[QNN_NQ] = (float)ev;
}

// ---------------------------------------------------------------------------
extern "C" void kernel_launch(void* const* d_in, const int* in_sizes, int n_in,
                              void* d_out, int out_size, void* d_ws, size_t ws_size,
                              hipStream_t stream) {
  (void)in_sizes; (void)n_in; (void)out_size; (void)d_ws; (void)ws_size;
  const float* x  = (const float*)d_in[0];   // (1024,)
  const float* W  = (const float*)d_in[1];   // (23, 1024)
  const float* qp = (const float*)d_in[2];   // (92,)
  float* out = (float*)d_out;                // 23 x h  +  1 x exp_val

  qnn_matvec_wmma<<<1, 64, 0, stream>>>(x, W, out);
  qnn_expval<<<1, 32, 0, stream>>>(qp, out);
}